// ClusterGCNConvEncoder_4801773437672
// MI455X (gfx1250) — compile-verified
//
#include <hip/hip_runtime.h>

// ---------------------------------------------------------------------------
// ClusterGCN 2-layer forward for MI455X (gfx1250, wave32).
// Dense parts use native fp32 WMMA: v_wmma_f32_16x16x4_f32.
// Scatter parts use float4 global loads + global_atomic_add_f32.
// ---------------------------------------------------------------------------

typedef float v2f __attribute__((ext_vector_type(2)));
typedef float v8f __attribute__((ext_vector_type(8)));

#define DFEAT 128   // D_IN == D_H == 128
#define DOUT2 16

// ----------------------------- utility kernels -----------------------------

__global__ void zero_f32_kernel(float* __restrict__ p, int n) {
    int i = blockIdx.x * blockDim.x + threadIdx.x;
    if (i < n) p[i] = 0.0f;
}

__global__ void edge_degree_kernel(const int* __restrict__ row,
                                   const int* __restrict__ col,
                                   float* __restrict__ deg, int e_cnt) {
    int e = blockIdx.x * blockDim.x + threadIdx.x;
    if (e >= e_cnt) return;
    int r = row[e];
    int c = col[e];
    if (r != c) atomicAdd(&deg[c], 1.0f);
}

__global__ void degree_inv_kernel(const float* __restrict__ deg,
                                  float* __restrict__ deg_inv, int n) {
    int i = blockIdx.x * blockDim.x + threadIdx.x;
    if (i < n) {
        float d = deg[i] + 1.0f;           // + self loop; always >= 1
        deg_inv[i] = 1.0f / d;
    }
}

// agg[i,:] = src[i,:] * deg_inv[i]   (self-loop term doubles as initializer)
__global__ void init_self_kernel(const float* __restrict__ src,
                                 const float* __restrict__ deg_inv,
                                 float* __restrict__ agg, int total) {
    int i = blockIdx.x * blockDim.x + threadIdx.x;
    if (i >= total) return;
    int node = i >> 7;                     // / DFEAT
    agg[i] = src[i] * deg_inv[node];
}

// One wave32 per edge; each lane moves 4 floats (float4 = global_load_b128).
__global__ void scatter_edges_kernel(const float* __restrict__ src,
                                     const int* __restrict__ row,
                                     const int* __restrict__ col,
                                     const float* __restrict__ deg_inv,
                                     float* __restrict__ agg, int e_cnt) {
    int gid  = blockIdx.x * blockDim.x + threadIdx.x;
    int e    = gid >> 5;
    int lane = gid & 31;
    if (e >= e_cnt) return;
    if (e + 8 < e_cnt) {                   // prefetch upcoming edge indices
        __builtin_prefetch(&row[e + 8], 0, 0);
        __builtin_prefetch(&col[e + 8], 0, 0);
    }
    int r = row[e];
    int c = col[e];
    if (r == c) return;                    // removed self loops
    float w = deg_inv[c];
    const float4 v = *reinterpret_cast<const float4*>(src + (size_t)r * DFEAT + (lane << 2));
    float* dst = agg + (size_t)c * DFEAT + (lane << 2);
    atomicAdd(dst + 0, v.x * w);
    atomicAdd(dst + 1, v.y * w);
    atomicAdd(dst + 2, v.z * w);
    atomicAdd(dst + 3, v.w * w);
}

// --------------------------- dual-input WMMA GEMM --------------------------
// out[tile 16x16] = A1@W1 + A2@W2 + bias  (optionally ReLU)
// A1,A2: [nrows,128] row-major.  W1,W2: [128,N_OUT] row-major.  K = 128.
// One wave per 16x16 output tile using v_wmma_f32_16x16x4_f32 (K-step 4).
//
// fp32 A 16x4 fragment (ISA 7.12.2): lane L (L<16) holds K={0,1} of row M=L,
// lane L+16 holds K={2,3}.  B 4x16 symmetric (row=K striped over VGPRs,
// N striped over lanes).  C/D: VGPR v -> M = v + 8*(lane>=16), N = lane&15.

template <int N_OUT, bool RELU>
__global__ void wmma_gemm_dual_kernel(const float* __restrict__ A1,
                                      const float* __restrict__ A2,
                                      const float* __restrict__ W1,
                                      const float* __restrict__ W2,
                                      const float* __restrict__ bias,
                                      float* __restrict__ out, int nrows) {
    constexpr int NT = N_OUT / 16;                     // N tiles
    int wave = (blockIdx.x * blockDim.x + threadIdx.x) >> 5;
    int lane = threadIdx.x & 31;
    int tm = wave / NT;
    int tn = wave % NT;
    if (tm * 16 >= nrows) return;                      // wave-uniform exit only

    const int mA   = lane & 15;                        // A row this lane feeds
    const int nB   = lane & 15;                        // B/C column this lane feeds
    const int koff = (lane >> 4) << 1;                 // 0 or 2

    const float* a1p = A1 + (size_t)(tm * 16 + mA) * DFEAT + koff;
    const float* a2p = A2 + (size_t)(tm * 16 + mA) * DFEAT + koff;
    const float* w1p = W1 + (size_t)koff * N_OUT + tn * 16 + nB;
    const float* w2p = W2 + (size_t)koff * N_OUT + tn * 16 + nB;

    v8f c = {};
#pragma unroll 4
    for (int k = 0; k < DFEAT; k += 4) {
        v2f a1, a2, b1, b2;
        a1.x = a1p[k];     a1.y = a1p[k + 1];
        b1.x = w1p[(size_t)k * N_OUT];
        b1.y = w1p[(size_t)(k + 1) * N_OUT];
        c = __builtin_amdgcn_wmma_f32_16x16x4_f32(false, a1, false, b1,
                                                  (short)0, c, false, false);
        a2.x = a2p[k];     a2.y = a2p[k + 1];
        b2.x = w2p[(size_t)k * N_OUT];
        b2.y = w2p[(size_t)(k + 1) * N_OUT];
        c = __builtin_amdgcn_wmma_f32_16x16x4_f32(false, a2, false, b2,
                                                  (short)0, c, false, false);
    }

    float bv = bias[tn * 16 + nB];
    int mbase = tm * 16 + ((lane >> 4) << 3);          // +8 for upper half-wave
#pragma unroll
    for (int v = 0; v < 8; ++v) {
        float val = c[v] + bv;
        if (RELU) val = fmaxf(val, 0.0f);
        out[(size_t)(mbase + v) * N_OUT + tn * 16 + nB] = val;
    }
}

// --------------------------------- launch ----------------------------------

extern "C" void kernel_launch(void* const* d_in, const int* in_sizes, int n_in,
                              void* d_out, int out_size, void* d_ws, size_t ws_size,
                              hipStream_t stream) {
    const float* x       = (const float*)d_in[0];
    const int*   eidx    = (const int*)d_in[1];
    const float* W_out1  = (const float*)d_in[2];
    const float* b_out1  = (const float*)d_in[3];
    const float* W_root1 = (const float*)d_in[4];
    const float* W_out2  = (const float*)d_in[5];
    const float* b_out2  = (const float*)d_in[6];
    const float* W_root2 = (const float*)d_in[7];
    float*       out     = (float*)d_out;

    const int N = in_sizes[0] / DFEAT;     // 100000 (multiple of 16)
    const int E = in_sizes[1] / 2;         // 640000
    const int* row = eidx;
    const int* col = eidx + E;

    // Workspace layout: deg | deg_inv | agg (reused both layers) | h
    float* deg  = (float*)d_ws;
    float* dinv = deg + N;
    float* agg  = dinv + N;
    float* h    = agg + (size_t)N * DFEAT;

    const int BT = 256;

    // --- degree / weights -------------------------------------------------
    zero_f32_kernel<<<(N + BT - 1) / BT, BT, 0, stream>>>(deg, N);
    edge_degree_kernel<<<(E + BT - 1) / BT, BT, 0, stream>>>(row, col, deg, E);
    degree_inv_kernel<<<(N + BT - 1) / BT, BT, 0, stream>>>(deg, dinv, N);

    const int total = N * DFEAT;
    const int scatter_blocks = (E * 32 + BT - 1) / BT;

    // --- layer 1: agg = x*dinv + scatter(x[row]*w) ------------------------
    init_self_kernel<<<(total + BT - 1) / BT, BT, 0, stream>>>(x, dinv, agg, total);
    scatter_edges_kernel<<<scatter_blocks, BT, 0, stream>>>(x, row, col, dinv, agg, E);

    // h = relu(agg@W_out1 + b1 + x@W_root1)   (N/16 * 128/16 tiles, 1 wave each)
    {
        long long tiles = (long long)(N / 16) * (DFEAT / 16);
        long long waves_per_block = BT / 32;
        int blocks = (int)((tiles + waves_per_block - 1) / waves_per_block);
        wmma_gemm_dual_kernel<DFEAT, true><<<blocks, BT, 0, stream>>>(
            agg, x, W_out1, W_root1, b_out1, h, N);
    }

    // --- layer 2: agg = h*dinv + scatter(h[row]*w) ------------------------
    init_self_kernel<<<(total + BT - 1) / BT, BT, 0, stream>>>(h, dinv, agg, total);
    scatter_edges_kernel<<<scatter_blocks, BT, 0, stream>>>(h, row, col, dinv, agg, E);

    // out = agg@W_out2 + b2 + h@W_root2   (N/16 tiles, D_OUT = 16)
    {
        long long tiles = (long long)(N / 16) * (DOUT2 / 16);
        long long waves_per_block = BT / 32;
        int blocks = (int)((tiles + waves_per_block - 1) / waves_per_block);
        wmma_gemm_dual_kernel<DOUT2, false><<<blocks, BT, 0, stream>>>(
            agg, h, W_out2, W_root2, b_out2, out, N);
    }
}